// RWKV_RNN_34265249087904
// MI455X (gfx1250) — compile-verified
//
#include <hip/hip_runtime.h>
#include <hip/hip_bf16.h>

// RWKV-5 WKV, chunked linear attention on CDNA5 fp32 WMMA (16x16x4).
// Grid: B*H*(S/16) = 512 blocks, 1 wave32 each; each wave owns a 16-column
// slice of the per-head 64x64 state (lives in LDS, fp32 end to end).
// v3: K-contiguous LDS layouts so every WMMA fragment is one aligned
// ds_load_b64; next-chunk k/r/v pipelined via global_prefetch (keeps VGPRs
// under 256, removes vgpr-msb / mov marshaling churn seen in v2).

#define B_ 4
#define T_ 4096
#define H_ 32
#define S_ 64
#define L_ 16
#define NCH_ (T_ / L_)
#define SCALE_ (1.0f / 128.0f)

typedef __attribute__((ext_vector_type(2))) float v2f;
typedef __attribute__((ext_vector_type(4))) float v4f;
typedef __attribute__((ext_vector_type(8))) float v8f;

#define KSTR 68  // float stride, 16x64 row-major chunk arrays and 16x64 transposed state
#define ASTR 18  // float stride, 16x16 masked-A (even -> 8B-aligned v2f frags)
#define HSTR 18  // float stride, 64x16 transposed K^ array

__launch_bounds__(32)
__global__ void rwkv5_wkv_chunk_kernel(
    const float* __restrict__ k, const float* __restrict__ v,
    const float* __restrict__ r, const float* __restrict__ s_in,
    const float* __restrict__ tf, const float* __restrict__ td,
    float* __restrict__ out, float* __restrict__ s_out)
{
    __shared__ float lds_rt[L_ * KSTR];           // R~[t][s] = r * w^t        (row-major)
    __shared__ float lds_kt[L_ * KSTR];           // K~[u][s] = k * w^-(u+1)   (row-major)
    __shared__ float lds_khT[S_ * HSTR];          // K^T[s][u] = k_u * w^(15-u) (s-major)
    __shared__ float lds_ST[16 * KSTR];           // state^T [col d][row s]
    __shared__ float lds_wpow[(L_ + 1) * KSTR];   // w^t,  t=0..16, per s
    __shared__ float lds_winv[(L_ + 1) * KSTR];   // w^-t, t=0..16, per s
    __shared__ float lds_wtf[S_];                 // w * time_first
    __shared__ float lds_diag[2 * L_];            // half-wave partials of bonus dot
    __shared__ float lds_A[L_ * ASTR];            // masked A [16][16]

    const int lane = threadIdx.x;
    const int lo = lane & 15;
    const int hi = lane >> 4;

    const int blk = blockIdx.x;
    const int db = blk & 3;             // which 16-column slice of d
    const int h  = (blk >> 2) & (H_ - 1);
    const int b  = blk >> 7;
    const int d0 = db * 16;

    // ---- per-head decay / bonus tables (w^t and w^-t for t=0..16) ----
    for (int ss = lane; ss < S_; ss += 32) {
        float w  = td[h * S_ + ss];
        float iw = 1.0f / w;
        float p = 1.0f, q = 1.0f;
        for (int t = 0; t <= L_; ++t) {
            lds_wpow[t * KSTR + ss] = p;
            lds_winv[t * KSTR + ss] = q;
            p *= w;
            q *= iw;
        }
        lds_wtf[ss] = w * tf[h * S_ + ss];
    }
    // ---- init transposed state slice from input state2 [B,H,S,S] ----
    {
        const float* sp = s_in + (((size_t)b * H_ + h) * S_) * S_ + d0;
        for (int i = 0; i < 32; ++i) {
            int row = i * 2 + hi;
            lds_ST[lo * KSTR + row] = sp[(size_t)row * S_ + lo];  // coalesced read
        }
    }
    __syncthreads();

    const size_t bh_off = ((size_t)b * T_) * H_ * S_ + (size_t)h * S_;

    // warm the cache for chunk 0
    {
        size_t g = bh_off + (size_t)(lane & 15) * H_ * S_ + (size_t)hi * 32;
        __builtin_prefetch(k + g, 0, 1);
        __builtin_prefetch(r + g, 0, 1);
        __builtin_prefetch(v + (bh_off + (size_t)(lane & 15) * H_ * S_ + d0), 0, 1);
    }

    for (int c = 0; c < NCH_; ++c) {
        const int t0 = c * L_;

        // ---- load chunk (cache-warm) + build pre-weighted LDS arrays ----
#pragma unroll
        for (int wv = 0; wv < 8; ++wv) {
            int linear = lane + wv * 32;
            int row = linear >> 4;            // token in chunk
            int q4  = (linear & 15) * 4;      // float offset in row
            size_t g = bh_off + (size_t)(t0 + row) * H_ * S_ + q4;
            v4f kk = *(const v4f*)(k + g);
            v4f rr = *(const v4f*)(r + g);
            v4f wp = *(const v4f*)(&lds_wpow[row * KSTR + q4]);
            v4f wi = *(const v4f*)(&lds_winv[(row + 1) * KSTR + q4]);
            v4f wh = *(const v4f*)(&lds_wpow[(15 - row) * KSTR + q4]);
            *(v4f*)(&lds_rt[row * KSTR + q4]) = rr * wp;
            *(v4f*)(&lds_kt[row * KSTR + q4]) = kk * wi;
            v4f khv = kk * wh;
#pragma unroll
            for (int j = 0; j < 4; ++j)
                lds_khT[(q4 + j) * HSTR + row] = khv[j];
        }

        // ---- prefetch next chunk's k/r/v lines behind this chunk's WMMAs ----
        if (c + 1 < NCH_) {
            int t0n = t0 + L_;
            size_t g = bh_off + (size_t)(t0n + lo) * H_ * S_ + (size_t)hi * 32;
            __builtin_prefetch(k + g, 0, 1);
            __builtin_prefetch(r + g, 0, 1);
            __builtin_prefetch(v + (bh_off + (size_t)(t0n + lo) * H_ * S_ + d0), 0, 1);
        }
        __syncthreads();

        // ---- V fragments (B operand, 4x16 sub-tiles over K=u) ----
        v2f vf[4];
#pragma unroll
        for (int kb = 0; kb < 4; ++kb) {
            int u0 = kb * 4 + 2 * hi;
            size_t g0 = bh_off + (size_t)(t0 + u0) * H_ * S_ + d0 + lo;
            vf[kb].x = v[g0];
            vf[kb].y = v[g0 + (size_t)H_ * S_];
        }

        // ---- bonus diag partials, all 32 lanes:
        //      r.k.tf = (R~ * K~) * (w*tf);  lane does 32-long partial ----
        {
            float part = 0.f;
            int sb = hi * 32;
#pragma unroll
            for (int i = 0; i < 32; ++i) {
                int ss = sb + i;
                part += lds_rt[lo * KSTR + ss] * lds_kt[lo * KSTR + ss] * lds_wtf[ss];
            }
            lds_diag[hi * L_ + lo] = part;
        }

        // ---- A = R~ @ K~^T : 16x16, K=64 (16 WMMAs, v2f frags = 1 b64 load) ----
        v8f accA = {};
        v2f rt[16];
#pragma unroll
        for (int kb = 0; kb < 16; ++kb) {
            int si = kb * 4 + 2 * hi;
            v2f a  = *(const v2f*)(&lds_rt[lo * KSTR + si]);
            v2f bb = *(const v2f*)(&lds_kt[lo * KSTR + si]);
            rt[kb] = a;
            accA = __builtin_amdgcn_wmma_f32_16x16x4_f32(
                false, a, false, bb, (short)0, accA, false, false);
        }

        // ---- causal mask; diagonal <- bonus term ----
#pragma unroll
        for (int i = 0; i < 8; ++i) {
            int row = 8 * hi + i;
            float val;
            if (lo < row)       val = accA[i];
            else if (lo == row) val = lds_diag[row] + lds_diag[L_ + row];
            else                val = 0.0f;
            lds_A[row * ASTR + lo] = val;
        }

        // ---- Out = R~ @ S0  +  A_mask @ V   (16+4 WMMAs) ----
        v8f acc = {};
#pragma unroll
        for (int kb = 0; kb < 16; ++kb) {
            int si = kb * 4 + 2 * hi;
            v2f bb = *(const v2f*)(&lds_ST[lo * KSTR + si]);   // state^T: K contiguous
            acc = __builtin_amdgcn_wmma_f32_16x16x4_f32(
                false, rt[kb], false, bb, (short)0, acc, false, false);
        }
#pragma unroll
        for (int kb = 0; kb < 4; ++kb) {
            int u0 = kb * 4 + 2 * hi;
            v2f a = *(const v2f*)(&lds_A[lo * ASTR + u0]);
            acc = __builtin_amdgcn_wmma_f32_16x16x4_f32(
                false, a, false, vf[kb], (short)0, acc, false, false);
        }

        // ---- store out chunk (scaled) ----
#pragma unroll
        for (int i = 0; i < 8; ++i) {
            int t = 8 * hi + i;
            out[bh_off + (size_t)(t0 + t) * H_ * S_ + d0 + lo] = SCALE_ * acc[i];
        }

        // ---- state update: S = diag(w^16) S + K^^T V  (4x4 WMMAs) ----
#pragma unroll
        for (int sb = 0; sb < 4; ++sb) {
            v8f st;
#pragma unroll
            for (int i = 0; i < 8; ++i) {
                int srow = sb * 16 + 8 * hi + i;
                st[i] = lds_ST[lo * KSTR + srow] * lds_wpow[16 * KSTR + srow];
            }
            int scol = sb * 16 + lo;
#pragma unroll
            for (int kb = 0; kb < 4; ++kb) {
                int u0 = kb * 4 + 2 * hi;
                v2f a = *(const v2f*)(&lds_khT[scol * HSTR + u0]);  // K^T: K contiguous
                st = __builtin_amdgcn_wmma_f32_16x16x4_f32(
                    false, a, false, vf[kb], (short)0, st, false, false);
            }
#pragma unroll
            for (int i = 0; i < 8; ++i) {
                int srow = sb * 16 + 8 * hi + i;
                lds_ST[lo * KSTR + srow] = st[i];
            }
        }
        __syncthreads();
    }

    // ---- write final state slice: s_final [B,H,S,S] ----
    {
        float* sp = s_out + (((size_t)b * H_ + h) * S_) * S_ + d0;
        for (int i = 0; i < 32; ++i) {
            int row = i * 2 + hi;
            sp[(size_t)row * S_ + lo] = lds_ST[lo * KSTR + row];  // coalesced write
        }
    }
}

extern "C" void kernel_launch(void* const* d_in, const int* in_sizes, int n_in,
                              void* d_out, int out_size, void* d_ws, size_t ws_size,
                              hipStream_t stream) {
    const float* k  = (const float*)d_in[0];
    const float* v  = (const float*)d_in[1];
    const float* r  = (const float*)d_in[2];
    const float* s2 = (const float*)d_in[3];
    const float* tf = (const float*)d_in[4];
    const float* td = (const float*)d_in[5];
    float* out  = (float*)d_out;
    float* sfin = out + (size_t)B_ * T_ * H_ * S_;

    dim3 grid(B_ * H_ * (S_ / 16));  // 512 independent scan chains
    dim3 block(32);                  // one wave32 per chain
    rwkv5_wkv_chunk_kernel<<<grid, block, 0, stream>>>(k, v, r, s2, tf, td, out, sfin);
}